// my_loss_43138651521376
// MI455X (gfx1250) — compile-verified
//
#include <hip/hip_runtime.h>
#include <math.h>

// ---------------------------------------------------------------------------
// Problem constants (match reference)
// ---------------------------------------------------------------------------
#define NROWS   262144
#define DIM     256
#define NCLUS   512
#define NTILES  (NCLUS / 16)          // 32 tiles of 16 centers
#define NPAIRS  ((NCLUS * (NCLUS - 1)) / 2)   // 130816

// Workspace layout (floats):
//   sums/centers : [0, 131072)
//   counts       : [131072, 131584)
//   accum        : [131584]
//   norms        : [131585, 132097)
#define WS_SUMS    0
#define WS_COUNTS  (NCLUS * DIM)          // 131072
#define WS_ACCUM   (WS_COUNTS + NCLUS)    // 131584
#define WS_NORMS   (WS_ACCUM + 1)         // 131585
#define WS_ZERO_N  (WS_ACCUM + 1)         // zero sums+counts+accum (131585)

typedef __attribute__((ext_vector_type(2))) float v2f;
typedef __attribute__((ext_vector_type(8))) float v8f;

// ---------------------------------------------------------------------------
// Kernel 1: zero sums + counts + accum (ws is poisoned, not re-zeroed by host)
// ---------------------------------------------------------------------------
__global__ void k_zero(float* __restrict__ ws, int n) {
    int t = blockIdx.x * blockDim.x + threadIdx.x;
    if (t < n) ws[t] = 0.0f;
}

// ---------------------------------------------------------------------------
// Kernel 2: scatter-add rows into cluster sums (f32 atomics resolved in L2).
// 64 consecutive lanes cover one 1KB row -> fully coalesced float4 loads.
// ---------------------------------------------------------------------------
__global__ void k_scatter(const float* __restrict__ mat,
                          const int*   __restrict__ label,
                          float* __restrict__ sums,
                          float* __restrict__ counts) {
    long long t = (long long)blockIdx.x * blockDim.x + threadIdx.x;
    int row = (int)(t >> 6);        // 64 float4-chunks per row
    int c4  = (int)(t & 63);
    if (row >= NROWS) return;
    int lab = label[row];
    const float4 v = ((const float4*)(mat + (size_t)row * DIM))[c4];
    float* base = sums + (size_t)lab * DIM + c4 * 4;
    atomicAdd(base + 0, v.x);
    atomicAdd(base + 1, v.y);
    atomicAdd(base + 2, v.z);
    atomicAdd(base + 3, v.w);
    if (c4 == 0) atomicAdd(counts + lab, 1.0f);
}

// ---------------------------------------------------------------------------
// Kernel 3: centers = sums / max(count,1) (in place) + squared norms.
// ---------------------------------------------------------------------------
__global__ void k_centers(float* __restrict__ sums,
                          const float* __restrict__ counts,
                          float* __restrict__ norms) {
    int c = blockIdx.x;       // 512 blocks
    int d = threadIdx.x;      // 256 threads
    float cnt = counts[c];
    float denom = cnt > 1.0f ? cnt : 1.0f;
    float v = sums[(size_t)c * DIM + d] / denom;
    sums[(size_t)c * DIM + d] = v;

    __shared__ float red[DIM];
    red[d] = v * v;
    __syncthreads();
    for (int s = DIM / 2; s > 0; s >>= 1) {
        if (d < s) red[d] += red[d + s];
        __syncthreads();
    }
    if (d == 0) norms[c] = red[0];
}

// ---------------------------------------------------------------------------
// Kernel 4: pdist via fp32 WMMA Gram tiles.
// One wave (32 lanes) per 16x16 tile-pair (I<=J); 64 x V_WMMA_F32_16X16X4_F32
// accumulate G = C_I . C_J^T over K=256. dist^2 = n_i + n_j - 2 g_ij.
// ---------------------------------------------------------------------------
__global__ __launch_bounds__(32)
void k_pdist(const float* __restrict__ centers,
             const float* __restrict__ norms,
             float* __restrict__ accum) {
    const int I = blockIdx.x;
    const int J = blockIdx.y;
    if (I > J) return;                         // uniform per wave: EXEC stays all-1s

    const int lane = threadIdx.x;              // 0..31 (wave32)
    const int half = lane >> 4;                // K-half selector
    const int lm   = lane & 15;                // M (for A) / N (for B)

    // 32-bit operand layout: lanes 0-15 carry K={0,1}, lanes 16-31 carry K={2,3}
    const float* arow = centers + (size_t)(I * 16 + lm) * DIM + 2 * half;
    const float* brow = centers + (size_t)(J * 16 + lm) * DIM + 2 * half;

    v8f acc = {};
#pragma unroll 16
    for (int k = 0; k < DIM; k += 4) {
        v2f a = *(const v2f*)(arow + k);
        v2f b = *(const v2f*)(brow + k);
        // (neg_a, A, neg_b, B, c_mod, C, reuse_a, reuse_b)
        acc = __builtin_amdgcn_wmma_f32_16x16x4_f32(
            false, a, false, b, (short)0, acc, false, false);
    }

    // C/D layout: VGPR r -> M = r + 8*half ; lane%16 -> N
    const float nj = norms[J * 16 + lm];
    float s = 0.0f;
#pragma unroll
    for (int r = 0; r < 8; ++r) {
        int gi = I * 16 + r + 8 * half;
        int gj = J * 16 + lm;
        if (gi < gj) {                         // upper triangle (handles diagonal tiles)
            float sq = norms[gi] + nj - 2.0f * acc[r];
            sq = sq > 1e-12f ? sq : 1e-12f;
            s += sqrtf(sq);
        }
    }
    atomicAdd(accum, s);
}

// ---------------------------------------------------------------------------
// Kernel 5: out = -mean(dist)
// ---------------------------------------------------------------------------
__global__ void k_finalize(const float* __restrict__ accum, float* __restrict__ out) {
    if (threadIdx.x == 0 && blockIdx.x == 0)
        out[0] = -accum[0] / (float)NPAIRS;
}

// ---------------------------------------------------------------------------
extern "C" void kernel_launch(void* const* d_in, const int* in_sizes, int n_in,
                              void* d_out, int out_size, void* d_ws, size_t ws_size,
                              hipStream_t stream) {
    const float* mat   = (const float*)d_in[0];
    const int*   label = (const int*)  d_in[1];
    float* ws     = (float*)d_ws;
    float* sums   = ws + WS_SUMS;     // becomes centers in place
    float* counts = ws + WS_COUNTS;
    float* accum  = ws + WS_ACCUM;
    float* norms  = ws + WS_NORMS;
    float* out    = (float*)d_out;

    // 1) zero accumulators
    k_zero<<<(WS_ZERO_N + 255) / 256, 256, 0, stream>>>(ws, WS_ZERO_N);

    // 2) scatter-add 256 MB of rows into 512 KB of sums (L2-resident atomics)
    long long work = (long long)NROWS * (DIM / 4);
    k_scatter<<<(unsigned)((work + 255) / 256), 256, 0, stream>>>(mat, label, sums, counts);

    // 3) normalize to centers + squared norms
    k_centers<<<NCLUS, DIM, 0, stream>>>(sums, counts, norms);

    // 4) pairwise distances via fp32 WMMA Gram tiles
    dim3 grid(NTILES, NTILES);
    k_pdist<<<grid, 32, 0, stream>>>(sums, norms, accum);

    // 5) finalize scalar
    k_finalize<<<1, 1, 0, stream>>>(accum, out);
}